// Layer_63350767616119
// MI455X (gfx1250) — compile-verified
//
#include <hip/hip_runtime.h>
#include <cstdint>
#include <cstddef>

// ---------------------------------------------------------------------------
// Problem dimensions (from reference): x is [T, N] f32, outputs are 4x [T, N].
// ---------------------------------------------------------------------------
#define T_DIM 2048
#define N_DIM 8192

// Chunked two-pass decomposition.
#define CHUNK 64                         // time-steps per pass-2 chunk
#define S_CHUNKS (T_DIM / CHUNK)         // 32 chunks

// Pass 1 (checkpoint scan): 128 neurons/block, 8-deep async LDS ring buffer.
#define P1_THREADS 128
#define P1_NBLK (N_DIM / P1_THREADS)     // 64 blocks
#define P1_R 32                          // rows (time-steps) per stage
#define P1_NBUF 8                        // ring-buffer depth
#define P1_STAGES (T_DIM / P1_R)         // 64 stages
#define P1_ISSUES (P1_R / 4)             // 8 b128 issues per thread per stage
#define P1_TILE_FLOATS (P1_R * P1_THREADS)          // 4096 floats = 16 KB
#define P1_LDS_BYTES (P1_NBUF * P1_TILE_FLOATS * 4) // 128 KB (<= 320 KB/WGP)

// Pass 2 (parallel replay): 256 neurons/block, one-shot 64 KB tile.
#define P2_THREADS 256
#define P2_NBLKX (N_DIM / P2_THREADS)    // 32
#define P2_ISSUES (CHUNK / 4)            // 16 b128 issues per thread
#define P2_LDS_BYTES (CHUNK * P2_THREADS * 4)       // 64 KB

// ---------------------------------------------------------------------------
// CDNA5 async global->LDS path (guarded; falls back to sync copy if absent).
// Builtin signature (from hipcc diagnostic):
//   void __builtin_amdgcn_global_load_async_to_lds_b128(
//       v4i addrspace(1)* gaddr, v4i addrspace(3)* lds, imm int off, imm int cpol)
// ---------------------------------------------------------------------------
typedef int v4i_t __attribute__((vector_size(4 * sizeof(int))));

#if defined(__has_builtin)
#  if __has_builtin(__builtin_amdgcn_global_load_async_to_lds_b128)
#    define HAVE_ASYNC_LDS 1
#  endif
#endif
#ifndef HAVE_ASYNC_LDS
#  define HAVE_ASYNC_LDS 0
#endif

#if defined(__has_builtin) && __has_builtin(__builtin_amdgcn_s_wait_asynccnt)
#  define WAIT_ASYNC(n) __builtin_amdgcn_s_wait_asynccnt(n)
#elif HAVE_ASYNC_LDS
#  define WAIT_ASYNC(n) asm volatile("s_wait_asynccnt %0" :: "i"(n) : "memory")
#else
#  define WAIT_ASYNC(n) ((void)0)
#endif

__device__ __forceinline__ void async_copy16(const float* g, float* l) {
#if HAVE_ASYNC_LDS
  // global_load_async_to_lds_b128: fire-and-forget 16B copy, tracked by ASYNCcnt.
  __builtin_amdgcn_global_load_async_to_lds_b128(
      (__attribute__((address_space(1))) v4i_t*)g,
      (__attribute__((address_space(3))) v4i_t*)l,
      /*offset=*/0, /*cpol=*/0);
#else
  float4 v = *(const float4*)g;
  *(float4*)l = v;
#endif
}

// ---------------------------------------------------------------------------
// LIF recurrence: exact transcription of the reference step() body.
//   reset computed from PREVIOUS mem (strict >), mem zeroed on reset,
//   spike from NEW mem; first lif1 call uses -0.5*spk2, its spike discarded.
// ---------------------------------------------------------------------------
__device__ __forceinline__ void snn_step(float x, float& mem1, float& mem2,
                                         float& spk2, float& spk1) {
  float k1 = (mem1 > 1.0f) ? 0.0f : 1.0f;
  mem1 = (0.9f * mem1 - 0.5f * spk2) * k1;
  float k2 = (mem1 > 1.0f) ? 0.0f : 1.0f;
  mem1 = (0.9f * mem1 + x) * k2;
  spk1 = (mem1 > 1.0f) ? 1.0f : 0.0f;
  float k3 = (mem2 > 1.0f) ? 0.0f : 1.0f;
  mem2 = (0.9f * mem2 + spk1) * k3;
  spk2 = (mem2 > 1.0f) ? 1.0f : 0.0f;
}

// ---------------------------------------------------------------------------
// Pass 1: full sequential scan keeping state only; checkpoints every CHUNK.
// Input streamed through an 8-deep async LDS ring buffer for latency hiding.
// ---------------------------------------------------------------------------
__device__ __forceinline__ void p1_issue_stage(const float* __restrict__ x,
                                               float* __restrict__ smem,
                                               int s, int n0, int tid) {
  float* dst = smem + (s % P1_NBUF) * P1_TILE_FLOATS;
  const float* src = x + (size_t)(s * P1_R) * N_DIM + n0;
#pragma unroll
  for (int k = 0; k < P1_ISSUES; ++k) {
    int flat16 = tid + k * P1_THREADS;   // 16-byte units within the tile
    int row    = flat16 >> 5;            // 32 units (128 floats) per row
    int col4   = (flat16 & 31) * 4;      // float index within row
    async_copy16(src + (size_t)row * N_DIM + col4, dst + flat16 * 4);
  }
}

__global__ void __launch_bounds__(P1_THREADS)
snn_pass1(const float* __restrict__ x, float* __restrict__ ws) {
  extern __shared__ float smem[];
  const int tid = threadIdx.x;
  const int n0  = blockIdx.x * P1_THREADS;
  const int n   = n0 + tid;

  // Prologue: prime NBUF-1 stages of the ring buffer.
#pragma unroll
  for (int s = 0; s < P1_NBUF - 1; ++s) p1_issue_stage(x, smem, s, n0, tid);

  float mem1 = 0.0f, mem2 = 0.0f, spk2 = 0.0f;

  for (int s = 0; s < P1_STAGES; ++s) {
    if (s + P1_NBUF - 1 < P1_STAGES) {
      p1_issue_stage(x, smem, s + P1_NBUF - 1, n0, tid);
      WAIT_ASYNC((P1_NBUF - 1) * P1_ISSUES);   // oldest stage complete (in-order)
    } else {
      WAIT_ASYNC(0);                           // tail: drain everything
    }
    __syncthreads();

    const float* buf = smem + (s % P1_NBUF) * P1_TILE_FLOATS;
#pragma unroll 4
    for (int r = 0; r < P1_R; ++r) {
      float spk1;
      snn_step(buf[r * P1_THREADS + tid], mem1, mem2, spk2, spk1);
    }
    __syncthreads();   // everyone done reading before this buffer is reused

    const int rows_done = (s + 1) * P1_R;
    if ((rows_done % CHUNK) == 0) {
      const int ci = rows_done / CHUNK;        // state entering chunk ci
      if (ci < S_CHUNKS) {
        ws[(size_t)ci * N_DIM + n]                  = mem1;
        ws[(size_t)(S_CHUNKS + ci) * N_DIM + n]     = mem2;
        ws[(size_t)(2 * S_CHUNKS + ci) * N_DIM + n] = spk2;
      }
    }
  }
}

// ---------------------------------------------------------------------------
// Pass 2: each block replays one 64-step chunk for 256 neurons from its
// checkpoint, staging x through LDS via async loads, streaming outputs with
// non-temporal stores. 1024 blocks / 8192 waves => store-bandwidth bound.
// ---------------------------------------------------------------------------
__global__ void __launch_bounds__(P2_THREADS)
snn_pass2(const float* __restrict__ x, const float* __restrict__ ws,
          float* __restrict__ spk1_o, float* __restrict__ mem1_o,
          float* __restrict__ spk2_o, float* __restrict__ mem2_o) {
  extern __shared__ float smem[];
  const int tid = threadIdx.x;
  const int n0  = blockIdx.x * P2_THREADS;
  const int n   = n0 + tid;
  const int ci  = blockIdx.y;
  const int t0  = ci * CHUNK;

  const float* src = x + (size_t)t0 * N_DIM + n0;
#pragma unroll
  for (int k = 0; k < P2_ISSUES; ++k) {
    int flat16 = tid + k * P2_THREADS;   // 16-byte units
    int row    = flat16 >> 6;            // 64 units (256 floats) per row
    int col4   = (flat16 & 63) * 4;
    async_copy16(src + (size_t)row * N_DIM + col4, smem + flat16 * 4);
  }

  float mem1 = 0.0f, mem2 = 0.0f, spk2 = 0.0f;
  if (ci > 0) {   // chunk 0 starts from zero state; others from checkpoints
    mem1 = ws[(size_t)ci * N_DIM + n];
    mem2 = ws[(size_t)(S_CHUNKS + ci) * N_DIM + n];
    spk2 = ws[(size_t)(2 * S_CHUNKS + ci) * N_DIM + n];
  }

  WAIT_ASYNC(0);
  __syncthreads();

#pragma unroll 4
  for (int t = 0; t < CHUNK; ++t) {
    float spk1;
    snn_step(smem[t * P2_THREADS + tid], mem1, mem2, spk2, spk1);
    const size_t idx = (size_t)(t0 + t) * N_DIM + n;
    __builtin_nontemporal_store(spk1, spk1_o + idx);
    __builtin_nontemporal_store(mem1, mem1_o + idx);
    __builtin_nontemporal_store(spk2, spk2_o + idx);
    __builtin_nontemporal_store(mem2, mem2_o + idx);
  }
}

// ---------------------------------------------------------------------------
// Fallback: single-pass scan (used only if the workspace is too small).
// ---------------------------------------------------------------------------
__global__ void __launch_bounds__(256)
snn_mono(const float* __restrict__ x,
         float* __restrict__ spk1_o, float* __restrict__ mem1_o,
         float* __restrict__ spk2_o, float* __restrict__ mem2_o) {
  const int n = blockIdx.x * blockDim.x + threadIdx.x;
  float mem1 = 0.0f, mem2 = 0.0f, spk2 = 0.0f;
  for (int t = 0; t < T_DIM; ++t) {
    const size_t idx = (size_t)t * N_DIM + n;
    float spk1;
    snn_step(__builtin_nontemporal_load(x + idx), mem1, mem2, spk2, spk1);
    __builtin_nontemporal_store(spk1, spk1_o + idx);
    __builtin_nontemporal_store(mem1, mem1_o + idx);
    __builtin_nontemporal_store(spk2, spk2_o + idx);
    __builtin_nontemporal_store(mem2, mem2_o + idx);
  }
}

// ---------------------------------------------------------------------------
extern "C" void kernel_launch(void* const* d_in, const int* in_sizes, int n_in,
                              void* d_out, int out_size, void* d_ws, size_t ws_size,
                              hipStream_t stream) {
  (void)in_sizes; (void)n_in; (void)out_size;
  const float* x = (const float*)d_in[0];
  float* out = (float*)d_out;
  const size_t TN = (size_t)T_DIM * N_DIM;
  float* spk1_o = out;
  float* mem1_o = out + TN;
  float* spk2_o = out + 2 * TN;
  float* mem2_o = out + 3 * TN;

  const size_t ws_need = (size_t)3 * S_CHUNKS * N_DIM * sizeof(float);  // 3 MB
  if (ws_size >= ws_need) {
    float* ws = (float*)d_ws;
    snn_pass1<<<P1_NBLK, P1_THREADS, P1_LDS_BYTES, stream>>>(x, ws);
    snn_pass2<<<dim3(P2_NBLKX, S_CHUNKS), P2_THREADS, P2_LDS_BYTES, stream>>>(
        x, ws, spk1_o, mem1_o, spk2_o, mem2_o);
  } else {
    snn_mono<<<N_DIM / 256, 256, 0, stream>>>(x, spk1_o, mem1_o, spk2_o, mem2_o);
  }
}